// DC_Net_56856777064808
// MI455X (gfx1250) — compile-verified
//
#include <hip/hip_runtime.h>
#include <hip/hip_bf16.h>

// ---------------------------------------------------------------------------
// DGCNN-style point cloud network for MI455X (gfx1250), fp32 throughout.
// Matrix work (KNN Gram tiles + all 1x1-conv GEMMs) runs on
// V_WMMA_F32_16X16X4_F32 (wave32). GEMM waves are register-blocked 64x16
// (4 accumulators sharing one B fragment per K-step).
// ---------------------------------------------------------------------------

typedef __attribute__((ext_vector_type(2))) float v2f;
typedef __attribute__((ext_vector_type(8))) float v8f;

constexpr int   BATCH = 2;
constexpr int   NPT   = 4096;             // points per batch
constexpr int   KNN   = 20;               // neighbors
constexpr int   NPTS  = BATCH * NPT;      // 8192 point-columns
constexpr int   NE    = NPTS * KNN;       // 163840 edge-columns
constexpr float RSQ   = 0.9999950000374997f; // 1/sqrt(1+1e-5)

__device__ __forceinline__ float lrelu(float v) { return v >= 0.f ? v : 0.2f * v; }

// ---------------------------------------------------------------------------
// Fused GEMM:  OUT(O,NCOL) = lrelu( (W(O,C) @ IN(C,NCOL)) * g + b )
// One wave per 64x16 output strip (4 WMMA accumulators, O % 64 == 0).
// A-frag: lane(0..15) holds row M=lane, K pair {kc,kc+1}; lanes 16..31 K pair
// {kc+2,kc+3}.  B-frag mirrors with N=lane&15.  C/D: vgpr r -> M = r + 8*half.
// ---------------------------------------------------------------------------
__global__ __launch_bounds__(32)
void wmma_gemm_bn_act(const float* __restrict__ W, const float* __restrict__ IN,
                      float* __restrict__ OUT, const float* __restrict__ BN,
                      int O, int C, int NCOL)
{
    const int lane = threadIdx.x;
    const int lr   = lane & 15;
    const int half = lane >> 4;
    const int col  = blockIdx.x * 16 + lr;   // output column
    const int mrow = blockIdx.y * 64;        // first output row of 64-row strip

    v8f acc0 = {}, acc1 = {}, acc2 = {}, acc3 = {};
    const float* w0 = W + (size_t)(mrow + lr) * C;

    for (int kc = 0; kc < C; kc += 4) {
        const int ka = kc + 2 * half;
        v2f b;
        b.x = IN[(size_t)(ka)     * NCOL + col];
        b.y = IN[(size_t)(ka + 1) * NCOL + col];
        const v2f a0 = *(const v2f*)(w0 + ka);
        const v2f a1 = *(const v2f*)(w0 + (size_t)16 * C + ka);
        const v2f a2 = *(const v2f*)(w0 + (size_t)32 * C + ka);
        const v2f a3 = *(const v2f*)(w0 + (size_t)48 * C + ka);
        if ((kc & 31) == 0 && kc + 32 < C)
            __builtin_prefetch(&IN[(size_t)(kc + 32) * NCOL + col], 0, 1);
        acc0 = __builtin_amdgcn_wmma_f32_16x16x4_f32(false, a0, false, b, (short)0, acc0, false, false);
        acc1 = __builtin_amdgcn_wmma_f32_16x16x4_f32(false, a1, false, b, (short)0, acc1, false, false);
        acc2 = __builtin_amdgcn_wmma_f32_16x16x4_f32(false, a2, false, b, (short)0, acc2, false, false);
        acc3 = __builtin_amdgcn_wmma_f32_16x16x4_f32(false, a3, false, b, (short)0, acc3, false, false);
    }

    v8f accs[4] = {acc0, acc1, acc2, acc3};
#pragma unroll
    for (int t = 0; t < 4; ++t) {
#pragma unroll
        for (int r = 0; r < 8; ++r) {
            const int orow = mrow + 16 * t + r + 8 * half;
            float v = accs[t][r] * (BN[orow] * RSQ) + BN[O + orow];
            OUT[(size_t)orow * NCOL + col] = lrelu(v);
        }
    }
}

// ---------------------------------------------------------------------------
// KNN with WMMA Gram tiles + register-resident top-20 insertion sort.
// One wave handles 16 query rows of one batch; sweeps all 4096 candidate
// columns in 16-wide tiles.  neg_d = 2*inner - xx_n - xx_m (self included,
// matching jax.lax.top_k on neg_d).
// ---------------------------------------------------------------------------
template <int CPAD>
__global__ __launch_bounds__(32)
void knn_topk(const float* __restrict__ X,   // (C, NPTS) layout
              const float* __restrict__ XX,  // (NPTS)
              int* __restrict__ IDX,         // (NE) neighbor index in [0,NPT)
              int Creal)
{
    const int lane = threadIdx.x;
    const int tilesPerBatch = NPT / 16;
    const int b    = blockIdx.x / tilesPerBatch;
    const int nb   = (blockIdx.x % tilesPerBatch) * 16;
    const int base = b * NPT;
    const int lr   = lane & 15;
    const int half = lane >> 4;

    __shared__ float ldsA[CPAD * 16];
    __shared__ float ldsB[CPAD * 16];
    __shared__ float tile[16 * 16];
    __shared__ float xxm[16];

    for (int t = lane; t < CPAD * 16; t += 32) {
        const int c = t >> 4, r = t & 15;
        ldsA[t] = (c < Creal) ? X[(size_t)c * NPTS + base + nb + r] : 0.f;
    }
    const float xxn = (lane < 16) ? XX[base + nb + lane] : 0.f;

    float hv[KNN];
    int   hi[KNN];
#pragma unroll
    for (int k = 0; k < KNN; ++k) { hv[k] = -3.0e38f; hi[k] = 0; }

    for (int mt = 0; mt < tilesPerBatch; ++mt) {
        const int mb = mt * 16;
        __syncthreads();
        for (int t = lane; t < CPAD * 16; t += 32) {
            const int c = t >> 4, j = t & 15;
            ldsB[t] = (c < Creal) ? X[(size_t)c * NPTS + base + mb + j] : 0.f;
        }
        if (lane < 16) xxm[lane] = XX[base + mb + lane];
        __syncthreads();

        v8f acc = {};
#pragma unroll 4
        for (int kc = 0; kc < CPAD; kc += 4) {
            const int ka = kc + 2 * half;
            v2f a, bfr;
            a.x   = ldsA[(ka)     * 16 + lr];
            a.y   = ldsA[(ka + 1) * 16 + lr];
            bfr.x = ldsB[(ka)     * 16 + lr];
            bfr.y = ldsB[(ka + 1) * 16 + lr];
            acc = __builtin_amdgcn_wmma_f32_16x16x4_f32(false, a, false, bfr,
                                                        (short)0, acc, false, false);
        }
#pragma unroll
        for (int r = 0; r < 8; ++r) tile[(r + 8 * half) * 16 + lr] = acc[r];
        __syncthreads();

        if (lane < 16) {
#pragma unroll
            for (int j = 0; j < 16; ++j) {
                const float v = 2.f * tile[lane * 16 + j] - xxn - xxm[j];
                const int   m = mb + j;
                if (v > hv[KNN - 1]) {
                    hv[KNN - 1] = v; hi[KNN - 1] = m;
#pragma unroll
                    for (int q = KNN - 1; q > 0; --q) {
                        if (hv[q] > hv[q - 1]) {
                            float tv = hv[q]; hv[q] = hv[q - 1]; hv[q - 1] = tv;
                            int   ti = hi[q]; hi[q] = hi[q - 1]; hi[q - 1] = ti;
                        }
                    }
                }
            }
        }
    }

    if (lane < 16) {
        const size_t p = (size_t)(base + nb + lane) * KNN;
#pragma unroll
        for (int k = 0; k < KNN; ++k) IDX[p + k] = hi[k];
    }
}

// ---------------------------------------------------------------------------
// Elementwise / glue kernels
// ---------------------------------------------------------------------------
__global__ void copy_x_kernel(const float* __restrict__ IN, float* __restrict__ OUT)
{
    const int t = blockIdx.x * blockDim.x + threadIdx.x;
    if (t >= 3 * NPTS) return;
    const int c = t / NPTS, p = t % NPTS, b = p / NPT, n = p % NPT;
    OUT[t] = IN[(size_t)b * 3 * NPT + (size_t)c * NPT + n];
}

__global__ void sqsum_kernel(const float* __restrict__ X, float* __restrict__ XX, int C)
{
    const int p = blockIdx.x * blockDim.x + threadIdx.x;
    if (p >= NPTS) return;
    float s = 0.f;
    for (int c = 0; c < C; ++c) { const float v = X[(size_t)c * NPTS + p]; s += v * v; }
    XX[p] = s;
}

// first edge-conv of a chain with Cin=3 (6 effective channels), O=64
__global__ void edge_conv6(const float* __restrict__ X, const int* __restrict__ IDX,
                           const float* __restrict__ W, const float* __restrict__ BN,
                           float* __restrict__ OUT)
{
    const int col = blockIdx.x * blockDim.x + threadIdx.x;
    if (col >= NE) return;
    const int p = col / KNN;
    const int b = p / NPT;
    const int j = IDX[col] + b * NPT;
    const float c0 = X[p], c1 = X[NPTS + p], c2 = X[2 * NPTS + p];
    const float d0 = X[j] - c0, d1 = X[NPTS + j] - c1, d2 = X[2 * NPTS + j] - c2;
#pragma unroll 4
    for (int o = 0; o < 64; ++o) {
        const float* w = W + o * 6;
        float v = w[0] * d0 + w[1] * d1 + w[2] * d2 + w[3] * c0 + w[4] * c1 + w[5] * c2;
        v = v * (BN[o] * RSQ) + BN[64 + o];
        OUT[(size_t)o * NE + col] = lrelu(v);
    }
}

// edge feature materialization for Cin=128: OUT(256, NE)
__global__ void gather_edge256(const float* __restrict__ X, const int* __restrict__ IDX,
                               float* __restrict__ OUT)
{
    const int col = blockIdx.x * blockDim.x + threadIdx.x;
    const int c   = blockIdx.y;
    if (col >= NE) return;
    const int p = col / KNN;
    const int b = p / NPT;
    float v;
    if (c < 128) {
        const int j = IDX[col] + b * NPT;
        v = X[(size_t)c * NPTS + j] - X[(size_t)c * NPTS + p];
    } else {
        v = X[(size_t)(c - 128) * NPTS + p];
    }
    OUT[(size_t)c * NE + col] = v;
}

// max over the 20 neighbors: IN(C,NE) -> OUT(C,NPTS) (optionally duplicated)
__global__ void maxk_kernel(const float* __restrict__ IN, float* __restrict__ OUT,
                            int C, int dup)
{
    const int t = blockIdx.x * blockDim.x + threadIdx.x;
    if (t >= C * NPTS) return;
    const int c = t / NPTS, p = t % NPTS;
    const float* src = IN + (size_t)c * NE + (size_t)p * KNN;
    float m = src[0];
#pragma unroll
    for (int k = 1; k < KNN; ++k) m = fmaxf(m, src[k]);
    OUT[(size_t)c * NPTS + p] = m;
    if (dup) OUT[(size_t)(c + C) * NPTS + p] = m;
}

// max over points per batch: IN(Cp,NPTS) -> OUT(B,Cp)
__global__ void pool_max_kernel(const float* __restrict__ IN, float* __restrict__ OUT, int Cp)
{
    const int t = blockIdx.x * blockDim.x + threadIdx.x;
    if (t >= BATCH * Cp) return;
    const int b = t / Cp, c = t % Cp;
    const float* src = IN + (size_t)c * NPTS + (size_t)b * NPT;
    float m = src[0];
    for (int n = 1; n < NPT; ++n) m = fmaxf(m, src[n]);
    OUT[t] = m;
}

// small FC:  mode 0 -> BN+lrelu,  mode 1 -> +bias (no act)
__global__ void fc_kernel(const float* __restrict__ IN, const float* __restrict__ W,
                          const float* __restrict__ BNB, float* __restrict__ OUT,
                          int Cin, int O, int mode)
{
    const int t = blockIdx.x * blockDim.x + threadIdx.x;
    if (t >= BATCH * O) return;
    const int b = t / O, o = t % O;
    const float* w  = W + (size_t)o * Cin;
    const float* in = IN + (size_t)b * Cin;
    float s = 0.f;
    for (int c = 0; c < Cin; ++c) s += w[c] * in[c];
    if (mode == 0) { s = s * (BNB[o] * RSQ) + BNB[O + o]; s = lrelu(s); }
    else           { s += BNB[o]; }
    OUT[t] = s;
}

// xt[b,d,n] = sum_c x[b,c,n] * t[b,c,d]
__global__ void transform_kernel(const float* __restrict__ X, const float* __restrict__ T,
                                 float* __restrict__ XT)
{
    const int t = blockIdx.x * blockDim.x + threadIdx.x;
    if (t >= 3 * NPTS) return;
    const int d = t / NPTS, p = t % NPTS, b = p / NPT;
    const float* tm = T + b * 9;
    XT[(size_t)d * NPTS + p] = X[p] * tm[0 * 3 + d] + X[NPTS + p] * tm[1 * 3 + d]
                             + X[2 * NPTS + p] * tm[2 * 3 + d];
}

// final 17-way logits + softmax -> OUT (B, N, 17)
__global__ void logits_softmax_kernel(const float* __restrict__ H, // (128, NPTS)
                                      const float* __restrict__ W, // (17, 128)
                                      float* __restrict__ OUT)
{
    const int p = blockIdx.x * blockDim.x + threadIdx.x;
    if (p >= NPTS) return;
    float l[17];
#pragma unroll
    for (int o = 0; o < 17; ++o) {
        const float* w = W + o * 128;
        float s = 0.f;
        for (int c = 0; c < 128; ++c) s += w[c] * H[(size_t)c * NPTS + p];
        l[o] = s;
    }
    float mx = l[0];
#pragma unroll
    for (int o = 1; o < 17; ++o) mx = fmaxf(mx, l[o]);
    float sum = 0.f;
#pragma unroll
    for (int o = 0; o < 17; ++o) { l[o] = __expf(l[o] - mx); sum += l[o]; }
    const float inv = 1.f / sum;
#pragma unroll
    for (int o = 0; o < 17; ++o) OUT[(size_t)p * 17 + o] = l[o] * inv;
}

// ---------------------------------------------------------------------------
// Workspace arena (float offsets)
// ---------------------------------------------------------------------------
constexpr size_t O_XBUF  = 0;
constexpr size_t O_XT    = O_XBUF  + (size_t)3 * NPTS;
constexpr size_t O_XX    = O_XT    + (size_t)3 * NPTS;
constexpr size_t O_IDX   = O_XX    + (size_t)NPTS;          // ints
constexpr size_t O_X3    = O_IDX   + (size_t)NE;
constexpr size_t O_X4    = O_X3    + (size_t)128 * NPTS;
constexpr size_t O_X5    = O_X4    + (size_t)128 * NPTS;
constexpr size_t O_CAT   = O_X5    + (size_t)64 * NPTS;
constexpr size_t O_T128  = O_CAT   + (size_t)320 * NPTS;
constexpr size_t O_POOL  = O_T128  + (size_t)128 * NPTS;
constexpr size_t O_FC1   = O_POOL  + (size_t)BATCH * 1024;
constexpr size_t O_FC2   = O_FC1   + (size_t)BATCH * 512;
constexpr size_t O_TMAT  = O_FC2   + (size_t)BATCH * 256;
constexpr size_t O_SBUF1 = O_TMAT  + 32;
constexpr size_t O_SBUF2 = O_SBUF1 + (size_t)64 * NE;
constexpr size_t O_EBIG  = O_SBUF2 + (size_t)64 * NE;       // 256*NE floats

extern "C" void kernel_launch(void* const* d_in, const int* in_sizes, int n_in,
                              void* d_out, int out_size, void* d_ws, size_t ws_size,
                              hipStream_t stream)
{
    (void)in_sizes; (void)n_in; (void)out_size; (void)ws_size;

    const float* x   = (const float*)d_in[0];
    const float* tw1 = (const float*)d_in[1];  const float* tb1 = (const float*)d_in[2];
    const float* tw2 = (const float*)d_in[3];  const float* tb2 = (const float*)d_in[4];
    const float* tw3 = (const float*)d_in[5];  const float* tb3 = (const float*)d_in[6];
    const float* tw4 = (const float*)d_in[7];  const float* tb4 = (const float*)d_in[8];
    const float* tl1 = (const float*)d_in[9];  const float* tb5 = (const float*)d_in[10];
    const float* tl2 = (const float*)d_in[11]; const float* tb6 = (const float*)d_in[12];
    const float* ttw = (const float*)d_in[13]; const float* ttb = (const float*)d_in[14];
    const float* w1  = (const float*)d_in[15]; const float* b1  = (const float*)d_in[16];
    const float* w2  = (const float*)d_in[17]; const float* b2  = (const float*)d_in[18];
    const float* w3  = (const float*)d_in[19]; const float* b3  = (const float*)d_in[20];
    const float* w4  = (const float*)d_in[21]; const float* b4  = (const float*)d_in[22];
    const float* w5  = (const float*)d_in[23]; const float* b5  = (const float*)d_in[24];
    const float* w6  = (const float*)d_in[25]; const float* b6  = (const float*)d_in[26];
    const float* w7  = (const float*)d_in[27]; const float* b7  = (const float*)d_in[28];
    const float* w8  = (const float*)d_in[29]; const float* b8  = (const float*)d_in[30];
    const float* w9  = (const float*)d_in[31]; const float* b9  = (const float*)d_in[32];
    const float* w10 = (const float*)d_in[33]; const float* b10 = (const float*)d_in[34];
    const float* w11 = (const float*)d_in[35]; const float* b11 = (const float*)d_in[36];
    const float* w12 = (const float*)d_in[37]; const float* b12 = (const float*)d_in[38];
    const float* w13 = (const float*)d_in[39]; const float* b13 = (const float*)d_in[40];
    const float* w14 = (const float*)d_in[41];

    float* ws    = (float*)d_ws;
    float* XBUF  = ws + O_XBUF;
    float* XT    = ws + O_XT;
    float* XX    = ws + O_XX;
    int*   IDX   = (int*)(ws + O_IDX);
    float* X3    = ws + O_X3;
    float* X4    = ws + O_X4;
    float* X5    = ws + O_X5;
    float* CAT   = ws + O_CAT;
    float* T128  = ws + O_T128;
    float* POOL  = ws + O_POOL;
    float* FC1   = ws + O_FC1;
    float* FC2   = ws + O_FC2;
    float* TMAT  = ws + O_TMAT;
    float* SBUF1 = ws + O_SBUF1;
    float* SBUF2 = ws + O_SBUF2;
    float* EBIG  = ws + O_EBIG;            // 256*NE; aliases below
    float* ELO   = EBIG;                   // (128, NE) tnet ping
    float* EHI   = EBIG + (size_t)128 * NE;// (128, NE) tnet pong
    float* TP1K  = EBIG;                   // (1024, NPTS) tnet pre-pool
    float* H1344 = EBIG;                   // (1344, NPTS) big concat

    const int knnBlocks = BATCH * (NPT / 16);
    auto gemm = [&](const float* W, const float* IN, float* OUT, const float* BN,
                    int O, int C, int NCOL) {
        wmma_gemm_bn_act<<<dim3(NCOL / 16, O / 64), 32, 0, stream>>>(W, IN, OUT, BN, O, C, NCOL);
    };

    // ---- phase 0: t-net -------------------------------------------------
    copy_x_kernel<<<(3 * NPTS + 255) / 256, 256, 0, stream>>>(x, XBUF);
    sqsum_kernel<<<(NPTS + 255) / 256, 256, 0, stream>>>(XBUF, XX, 3);
    knn_topk<4><<<knnBlocks, 32, 0, stream>>>(XBUF, XX, IDX, 3);
    edge_conv6<<<(NE + 255) / 256, 256, 0, stream>>>(XBUF, IDX, tw1, tb1, SBUF1);
    gemm(tw2, SBUF1, ELO, tb2, 128, 64, NE);
    gemm(tw3, ELO, EHI, tb3, 128, 128, NE);
    maxk_kernel<<<(128 * NPTS + 255) / 256, 256, 0, stream>>>(EHI, T128, 128, 0);
    gemm(tw4, T128, TP1K, tb4, 1024, 128, NPTS);
    pool_max_kernel<<<(BATCH * 1024 + 255) / 256, 256, 0, stream>>>(TP1K, POOL, 1024);
    fc_kernel<<<(BATCH * 512 + 255) / 256, 256, 0, stream>>>(POOL, tl1, tb5, FC1, 1024, 512, 0);
    fc_kernel<<<(BATCH * 256 + 255) / 256, 256, 0, stream>>>(FC1, tl2, tb6, FC2, 512, 256, 0);
    fc_kernel<<<1, 256, 0, stream>>>(FC2, ttw, ttb, TMAT, 256, 9, 1);
    transform_kernel<<<(3 * NPTS + 255) / 256, 256, 0, stream>>>(XBUF, TMAT, XT);

    // ---- phase 1: chain 1 (C=3 edge features) ---------------------------
    sqsum_kernel<<<(NPTS + 255) / 256, 256, 0, stream>>>(XT, XX, 3);
    knn_topk<4><<<knnBlocks, 32, 0, stream>>>(XT, XX, IDX, 3);
    edge_conv6<<<(NE + 255) / 256, 256, 0, stream>>>(XT, IDX, w1, b1, SBUF1);
    gemm(w2, SBUF1, SBUF2, b2, 64, 64, NE);
    gemm(w3, SBUF2, SBUF1, b3, 64, 64, NE);
    maxk_kernel<<<(64 * NPTS + 255) / 256, 256, 0, stream>>>(SBUF1, X3, 64, 1); // X3 = 128ch

    // ---- phase 2: chain 2 (C=128) ---------------------------------------
    sqsum_kernel<<<(NPTS + 255) / 256, 256, 0, stream>>>(X3, XX, 128);
    knn_topk<128><<<knnBlocks, 32, 0, stream>>>(X3, XX, IDX, 128);
    gather_edge256<<<dim3((NE + 255) / 256, 256), 256, 0, stream>>>(X3, IDX, EBIG);
    gemm(w4, EBIG, SBUF1, b4, 64, 256, NE);
    gemm(w5, SBUF1, SBUF2, b5, 64, 64, NE);
    gemm(w6, SBUF2, SBUF1, b6, 64, 64, NE);
    maxk_kernel<<<(64 * NPTS + 255) / 256, 256, 0, stream>>>(SBUF1, X4, 64, 1); // X4 = 128ch

    // ---- phase 3: chain 3 (C=128) ---------------------------------------
    sqsum_kernel<<<(NPTS + 255) / 256, 256, 0, stream>>>(X4, XX, 128);
    knn_topk<128><<<knnBlocks, 32, 0, stream>>>(X4, XX, IDX, 128);
    gather_edge256<<<dim3((NE + 255) / 256, 256), 256, 0, stream>>>(X4, IDX, EBIG);
    gemm(w7, EBIG, SBUF1, b7, 64, 256, NE);
    gemm(w8, SBUF1, SBUF2, b8, 64, 64, NE);
    gemm(w9, SBUF2, SBUF1, b9, 64, 64, NE);
    maxk_kernel<<<(64 * NPTS + 255) / 256, 256, 0, stream>>>(SBUF1, X5, 64, 0); // X5 = 64ch

    // ---- phase 4: global head -------------------------------------------
    hipMemcpyAsync(CAT,                       X3, (size_t)128 * NPTS * 4, hipMemcpyDeviceToDevice, stream);
    hipMemcpyAsync(CAT + (size_t)128 * NPTS,  X4, (size_t)128 * NPTS * 4, hipMemcpyDeviceToDevice, stream);
    hipMemcpyAsync(CAT + (size_t)256 * NPTS,  X5, (size_t)64  * NPTS * 4, hipMemcpyDeviceToDevice, stream);
    gemm(w10, CAT, H1344, b10, 1024, 320, NPTS);       // rows 0..1023 of H1344
    hipMemcpyAsync(H1344 + (size_t)1024 * NPTS, X3, (size_t)128 * NPTS * 4, hipMemcpyDeviceToDevice, stream);
    hipMemcpyAsync(H1344 + (size_t)1152 * NPTS, X4, (size_t)128 * NPTS * 4, hipMemcpyDeviceToDevice, stream);
    hipMemcpyAsync(H1344 + (size_t)1280 * NPTS, X5, (size_t)64  * NPTS * 4, hipMemcpyDeviceToDevice, stream);
    gemm(w11, H1344, SBUF1, b11, 256, 1344, NPTS);
    gemm(w12, SBUF1, SBUF2, b12, 256, 256, NPTS);
    gemm(w13, SBUF2, SBUF1, b13, 128, 256, NPTS);
    logits_softmax_kernel<<<(NPTS + 255) / 256, 256, 0, stream>>>(SBUF1, w14, (float*)d_out);
}